// multiMask_MixTemplateModel_52467320487997
// MI455X (gfx1250) — compile-verified
//
#include <hip/hip_runtime.h>
#include <math.h>

typedef float v8f __attribute__((ext_vector_type(8)));
typedef float v2f __attribute__((ext_vector_type(2)));

#define B_  256
#define M_  5
#define NR  1280          // B*M rows
#define D_  768
#define E_  8
#define V_  28996
#define NJ  26            // total verbalizer ids (5+6+5+5+5)
#define DROWS 32          // rows per dense workgroup

__device__ __forceinline__ v8f wmma4(v2f a, v2f b, v8f c) {
    // V_WMMA_F32_16X16X4_F32 : D = A(16x4 f32) * B(4x16 f32) + C(16x16 f32)
    return __builtin_amdgcn_wmma_f32_16x16x4_f32(
        /*neg_a=*/false, a, /*neg_b=*/false, b,
        /*c_mod=*/(short)0, c, /*reuse_a=*/false, /*reuse_b=*/false);
}

// ---------------------------------------------------------------------------
// K0: gather 26 decoder columns + biases per expert.
// dcol[e][j][d] = dec_w[e, d, id_j]; dbias[e][j] = dec_b[e, id_j]
// ---------------------------------------------------------------------------
__global__ __launch_bounds__(256) void gather_dec_kernel(
    const float* __restrict__ dec_w, const float* __restrict__ dec_b,
    const int* __restrict__ cpu, const int* __restrict__ graphic,
    const int* __restrict__ hardisk, const int* __restrict__ ram,
    const int* __restrict__ screen,
    float* __restrict__ dcol, float* __restrict__ dbias)
{
    int blk = blockIdx.x;           // 0..E_*NJ-1
    int e = blk / NJ, j = blk % NJ;
    int id;
    if      (j < 5)  id = cpu[j];
    else if (j < 11) id = graphic[j - 5];
    else if (j < 16) id = hardisk[j - 11];
    else if (j < 21) id = ram[j - 16];
    else             id = screen[j - 21];
    for (int d = threadIdx.x; d < D_; d += 256)
        dcol[((size_t)e * NJ + j) * D_ + d] =
            dec_w[((size_t)e * D_ + d) * V_ + id];
    if (threadIdx.x == 0)
        dbias[e * NJ + j] = dec_b[(size_t)e * V_ + id];
}

// ---------------------------------------------------------------------------
// K1: t = tanh(x @ gw1 + gb1)   [1280 x 768], f32 WMMA 16x16x4
// Workgroup = 64 rows x 128 cols; 8 waves, each wave 4 row-tiles x 1 col-tile
// ---------------------------------------------------------------------------
__global__ __launch_bounds__(256) void gate_h_kernel(
    const float* __restrict__ x, const float* __restrict__ gw1,
    const float* __restrict__ gb1, float* __restrict__ t)
{
    int tid  = threadIdx.x;
    int wave = tid >> 5;
    int lane = tid & 31;
    int l16  = lane & 15;
    int hi   = lane >> 4;                     // 0/1: K-half selector
    int colbase = blockIdx.x * 128 + wave * 16;
    int rowbase = blockIdx.y * 64;
    int col = colbase + l16;

    v8f acc[4] = {};
    const float* Arow0 = x + (size_t)(rowbase +  0 + l16) * D_;
    const float* Arow1 = x + (size_t)(rowbase + 16 + l16) * D_;
    const float* Arow2 = x + (size_t)(rowbase + 32 + l16) * D_;
    const float* Arow3 = x + (size_t)(rowbase + 48 + l16) * D_;

    for (int k = 0; k < D_; k += 4) {
        int kk = k + 2 * hi;
        if (k + 16 < D_)   // stream B panel ahead -> global_prefetch_b8
            __builtin_prefetch(&gw1[(size_t)(kk + 16) * D_ + col], 0, 1);
        v2f b;
        b.x = gw1[(size_t)kk * D_ + col];
        b.y = gw1[(size_t)(kk + 1) * D_ + col];
        v2f a0 = *(const v2f*)(Arow0 + kk);
        v2f a1 = *(const v2f*)(Arow1 + kk);
        v2f a2 = *(const v2f*)(Arow2 + kk);
        v2f a3 = *(const v2f*)(Arow3 + kk);
        acc[0] = wmma4(a0, b, acc[0]);
        acc[1] = wmma4(a1, b, acc[1]);
        acc[2] = wmma4(a2, b, acc[2]);
        acc[3] = wmma4(a3, b, acc[3]);
    }
    float bias = gb1[col];
#pragma unroll
    for (int rt = 0; rt < 4; ++rt) {
        int r0 = rowbase + rt * 16 + hi * 8;  // C layout: lanes>=16 hold M+8
#pragma unroll
        for (int vr = 0; vr < 8; ++vr)
            t[(size_t)(r0 + vr) * D_ + col] = tanhf(acc[rt][vr] + bias);
    }
}

// ---------------------------------------------------------------------------
// K2: g = t @ gw2 [1280 x 8]; log_softmax, probs, p*logp row-sum, top-3 gates
// One wave per row.
// ---------------------------------------------------------------------------
__global__ __launch_bounds__(256) void gate_logits_kernel(
    const float* __restrict__ t, const float* __restrict__ gw2,
    float* __restrict__ probs, float* __restrict__ rowplogp,
    float* __restrict__ topg, int* __restrict__ topi)
{
    int row  = blockIdx.x * 8 + (threadIdx.x >> 5);   // grid 160 * 8 waves = 1280
    int lane = threadIdx.x & 31;
    const float* tr = t + (size_t)row * D_;
    float acc[E_] = {};
    for (int d = lane; d < D_; d += 32) {
        float xv = tr[d];
        const float* w = gw2 + (size_t)d * E_;
#pragma unroll
        for (int e = 0; e < E_; ++e) acc[e] += xv * w[e];
    }
#pragma unroll
    for (int off = 16; off > 0; off >>= 1)
#pragma unroll
        for (int e = 0; e < E_; ++e)
            acc[e] += __shfl_down(acc[e], off, 32);

    if (lane == 0) {
        float mx = acc[0];
#pragma unroll
        for (int e = 1; e < E_; ++e) mx = fmaxf(mx, acc[e]);
        float se = 0.f;
#pragma unroll
        for (int e = 0; e < E_; ++e) se += expf(acc[e] - mx);
        float lse = logf(se);
        float p[E_], pl = 0.f;
#pragma unroll
        for (int e = 0; e < E_; ++e) {
            float lp = acc[e] - mx - lse;
            p[e] = expf(lp);
            probs[(size_t)row * E_ + e] = p[e];
            pl += p[e] * lp;
        }
        rowplogp[row] = pl;
        // top-3, ties -> lower index first (matches jax.lax.top_k)
        float b0 = -1.f, b1 = -1.f, b2 = -1.f; int j0 = 0, j1 = 0, j2 = 0;
#pragma unroll
        for (int e = 0; e < E_; ++e) {
            float v = p[e];
            if (v > b0)      { b2 = b1; j2 = j1; b1 = b0; j1 = j0; b0 = v; j0 = e; }
            else if (v > b1) { b2 = b1; j2 = j1; b1 = v;  j1 = e; }
            else if (v > b2) { b2 = v;  j2 = e; }
        }
        float s = 1.f / (b0 + b1 + b2 + 1e-6f);
        topg[row * 3 + 0] = b0 * s; topi[row * 3 + 0] = j0;
        topg[row * 3 + 1] = b1 * s; topi[row * 3 + 1] = j1;
        topg[row * 3 + 2] = b2 * s; topi[row * 3 + 2] = j2;
    }
}

// ---------------------------------------------------------------------------
// K5: miloss reduction -> d_out[6656]
// ---------------------------------------------------------------------------
__global__ __launch_bounds__(256) void miloss_kernel(
    const float* __restrict__ probs, const float* __restrict__ rowplogp,
    float* __restrict__ out_miloss)
{
    __shared__ float spe[256][E_];
    __shared__ float snp[256];
    int tid = threadIdx.x;
    float pe[E_] = {}; float np = 0.f;
    for (int r = tid; r < NR; r += 256) {
#pragma unroll
        for (int e = 0; e < E_; ++e) pe[e] += probs[(size_t)r * E_ + e];
        np += rowplogp[r];
    }
#pragma unroll
    for (int e = 0; e < E_; ++e) spe[tid][e] = pe[e];
    snp[tid] = np;
    __syncthreads();
    for (int off = 128; off > 0; off >>= 1) {
        if (tid < off) {
#pragma unroll
            for (int e = 0; e < E_; ++e) spe[tid][e] += spe[tid + off][e];
            snp[tid] += snp[tid + off];
        }
        __syncthreads();
    }
    if (tid == 0) {
        const float p_x = 1.0f / (float)NR;
        float He = 0.f;
#pragma unroll
        for (int e = 0; e < E_; ++e) {
            float v = spe[0][e] * p_x;
            He += v * logf(v);
        }
        out_miloss[0] = -(p_x * snp[0]) + He;
    }
}

// ---------------------------------------------------------------------------
// K3: dense expert pass, fully fused: h = LN(gelu(x @ dense_w[e] + b)) in LDS,
// then 26-column decoder dots. One WG = 32 rows x all 768 cols for expert e.
// 8 waves; each wave: 2 row-tiles x 6 col-tiles = 12 WMMA accumulators.
// Weight panel read once per 32 rows (halves L2 traffic vs 16-row tiles).
// ---------------------------------------------------------------------------
__global__ __launch_bounds__(256) void dense_kernel(
    const float* __restrict__ x, const float* __restrict__ dense_w,
    const float* __restrict__ dense_b, const float* __restrict__ ln_g,
    const float* __restrict__ ln_b, const float* __restrict__ dcol,
    const float* __restrict__ dbias, float* __restrict__ outall)
{
    __shared__ float hbuf[DROWS][772];     // 768 cols + pad, 16B-aligned rows
    __shared__ float redsum[DROWS][9];
    __shared__ float redsq[DROWS][9];
    __shared__ float mu_s[DROWS], rs_s[DROWS];

    int e       = blockIdx.y;
    int rowbase = blockIdx.x * DROWS;
    int tid  = threadIdx.x;
    int wave = tid >> 5;
    int lane = tid & 31;
    int l16  = lane & 15;
    int hi   = lane >> 4;
    int colb = wave * 96;

    const float* W     = dense_w + (size_t)e * D_ * D_;
    const float* Arow0 = x + (size_t)(rowbase + l16) * D_;
    const float* Arow1 = x + (size_t)(rowbase + 16 + l16) * D_;

    v8f acc[2][6] = {};
    for (int k = 0; k < D_; k += 4) {
        int kk = k + 2 * hi;
        v2f a0 = *(const v2f*)(Arow0 + kk);
        v2f a1 = *(const v2f*)(Arow1 + kk);
        if (k + 16 < D_)   // stream weight panel ahead -> global_prefetch_b8
            __builtin_prefetch(&W[(size_t)(kk + 16) * D_ + colb + l16], 0, 1);
#pragma unroll
        for (int ct = 0; ct < 6; ++ct) {
            int col = colb + ct * 16 + l16;
            v2f b;
            b.x = W[(size_t)kk * D_ + col];
            b.y = W[(size_t)(kk + 1) * D_ + col];
            acc[0][ct] = wmma4(a0, b, acc[0][ct]);
            acc[1][ct] = wmma4(a1, b, acc[1][ct]);
        }
    }
    // spill C tiles to LDS with bias
#pragma unroll
    for (int rt = 0; rt < 2; ++rt)
#pragma unroll
        for (int ct = 0; ct < 6; ++ct) {
            int col = colb + ct * 16 + l16;
            float bias = dense_b[(size_t)e * D_ + col];
#pragma unroll
            for (int vr = 0; vr < 8; ++vr)
                hbuf[rt * 16 + vr + hi * 8][col] = acc[rt][ct][vr] + bias;
        }
    __syncthreads();
    // exact GELU
    for (int i = tid; i < DROWS * D_; i += 256) {
        int r = i / D_, c = i % D_;
        float h = hbuf[r][c];
        hbuf[r][c] = 0.5f * h * (1.0f + erff(h * 0.70710678118654752f));
    }
    __syncthreads();
    // LayerNorm: 8 threads per row, 96 cols each
    {
        int row = tid >> 3, sub = tid & 7;
        float s = 0.f, sq = 0.f;
        for (int c = sub * 96; c < sub * 96 + 96; ++c) {
            float h = hbuf[row][c];
            s += h; sq += h * h;
        }
        redsum[row][sub] = s; redsq[row][sub] = sq;
    }
    __syncthreads();
    if ((tid & 7) == 0) {
        int row = tid >> 3;
        float s = 0.f, sq = 0.f;
        for (int i = 0; i < 8; ++i) { s += redsum[row][i]; sq += redsq[row][i]; }
        float mu  = s * (1.0f / (float)D_);
        float var = sq * (1.0f / (float)D_) - mu * mu;
        mu_s[row] = mu;
        rs_s[row] = rsqrtf(var + 1e-12f);
    }
    __syncthreads();
    {
        int row = tid >> 3, sub = tid & 7;
        float mu = mu_s[row], rs = rs_s[row];
        for (int c = sub * 96; c < sub * 96 + 96; ++c) {
            float h = hbuf[row][c];
            hbuf[row][c] = (h - mu) * rs * ln_g[(size_t)e * D_ + c]
                         + ln_b[(size_t)e * D_ + c];
        }
    }
    __syncthreads();
    // decoder: DROWS rows x 26 gathered vocab columns, dots of length 768
    for (int idx = tid; idx < DROWS * NJ; idx += 256) {
        int row = idx / NJ, j = idx % NJ;
        const float4* hv = (const float4*)hbuf[row];
        const float4* wv = (const float4*)(dcol + ((size_t)e * NJ + j) * D_);
        float sum = 0.f;
        for (int d = 0; d < D_ / 4; ++d) {
            float4 a = hv[d], b = wv[d];
            sum += a.x * b.x + a.y * b.y + a.z * b.z + a.w * b.w;
        }
        outall[((size_t)(rowbase + row) * E_ + e) * NJ + j] = sum + dbias[e * NJ + j];
    }
}

// ---------------------------------------------------------------------------
// K4: top-3 combine + verbalizer scatter into d_out
// ---------------------------------------------------------------------------
__global__ __launch_bounds__(256) void combine_kernel(
    const float* __restrict__ outall, const float* __restrict__ topg,
    const int* __restrict__ topi, float* __restrict__ out)
{
    int j = blockIdx.x;     // 0..25
    int b = threadIdx.x;    // 0..255
    int m, jl, base, n;
    if      (j < 5)  { m = 0; jl = j;      base = 0;    n = 5; }
    else if (j < 11) { m = 1; jl = j - 5;  base = 1280; n = 6; }
    else if (j < 16) { m = 2; jl = j - 11; base = 2816; n = 5; }
    else if (j < 21) { m = 3; jl = j - 16; base = 4096; n = 5; }
    else             { m = 4; jl = j - 21; base = 5376; n = 5; }
    int row = b * M_ + m;
    float v = 0.f;
#pragma unroll
    for (int k = 0; k < 3; ++k) {
        int e = topi[row * 3 + k];
        v += topg[row * 3 + k] * outall[((size_t)row * E_ + e) * NJ + j];
    }
    out[base + b * n + jl] = v;
}

// ---------------------------------------------------------------------------
extern "C" void kernel_launch(void* const* d_in, const int* in_sizes, int n_in,
                              void* d_out, int out_size, void* d_ws, size_t ws_size,
                              hipStream_t stream)
{
    const float* x       = (const float*)d_in[0];
    const float* gw1     = (const float*)d_in[1];
    const float* gb1     = (const float*)d_in[2];
    const float* gw2     = (const float*)d_in[3];
    const float* dense_w = (const float*)d_in[4];
    const float* dense_b = (const float*)d_in[5];
    const float* ln_g    = (const float*)d_in[6];
    const float* ln_b    = (const float*)d_in[7];
    const float* dec_w   = (const float*)d_in[8];
    const float* dec_b   = (const float*)d_in[9];
    const int* cpu_ids     = (const int*)d_in[10];
    const int* graphic_ids = (const int*)d_in[11];
    const int* hardisk_ids = (const int*)d_in[12];
    const int* ram_ids     = (const int*)d_in[13];
    const int* screen_ids  = (const int*)d_in[14];
    float* out = (float*)d_out;

    // workspace carving (all offsets multiples of 16 bytes)
    char* p = (char*)d_ws;
    float* t        = (float*)p; p += (size_t)NR * D_ * sizeof(float);   // 3.93 MB
    float* probs    = (float*)p; p += (size_t)NR * E_ * sizeof(float);
    float* rowplogp = (float*)p; p += (size_t)NR * sizeof(float);
    float* topg     = (float*)p; p += (size_t)NR * 3 * sizeof(float);
    int*   topi     = (int*)p;   p += (size_t)NR * 3 * sizeof(int);
    float* dcol     = (float*)p; p += (size_t)E_ * NJ * D_ * sizeof(float);
    float* dbias    = (float*)p; p += (size_t)E_ * NJ * sizeof(float) + 48; // pad
    float* outall   = (float*)p;                                            // 1.06 MB

    gather_dec_kernel<<<E_ * NJ, 256, 0, stream>>>(
        dec_w, dec_b, cpu_ids, graphic_ids, hardisk_ids, ram_ids, screen_ids,
        dcol, dbias);
    gate_h_kernel<<<dim3(D_ / 128, NR / 64), 256, 0, stream>>>(x, gw1, gb1, t);
    gate_logits_kernel<<<NR / 8, 256, 0, stream>>>(t, gw2, probs, rowplogp, topg, topi);
    miloss_kernel<<<1, 256, 0, stream>>>(probs, rowplogp, out + 6656);
    dense_kernel<<<dim3(NR / DROWS, E_), 256, 0, stream>>>(
        x, dense_w, dense_b, ln_g, ln_b, dcol, dbias, outall);
    combine_kernel<<<NJ, 256, 0, stream>>>(outall, topg, topi, out);
}